// FRAMES_VisionTransformer_28166395527587
// MI455X (gfx1250) — compile-verified
//
#include <hip/hip_runtime.h>
#include <stdint.h>

// Downsample-gather for the VisionTransformer volume op.
// One workgroup handles (b, o0, o1_half): stages a 98-token x 92-float window
// of the input into LDS via CDNA5 async global->LDS b128 loads, then scatters
// 32x64 outputs from LDS with coalesced stores.

#define TOK_ROWS 98            // 7 g0 values * 14 g1 values
#define CH_W     92            // aligned channel window (covers worst case 89)
#define VEC_PER_ROW (CH_W / 4) // 23 x b128 per row
#define TOT_VEC  (TOK_ROWS * VEC_PER_ROW)   // 2254 b128 loads per WG

__global__ __launch_bounds__(256)
void FRAMES_VisionTransformer_28166395527587_kernel(
    const float* __restrict__ hs,   // (64, 3137, 768) fp32
    float* __restrict__ out)        // (64, 1, 64, 64, 64) fp32
{
    const float R = (float)(768.0 / 729.0);   // exact f32 of the JAX scale

    const int wg   = blockIdx.x;         // [0, 64*64*2)
    const int half = wg & 1;             // which o1 half (g0 in [0,7) or [7,14))
    const int o0   = (wg >> 1) & 63;
    const int b    = wg >> 7;

    const int z0      = (o0 * 9) >> 2;   // floor(o0*2.25), exact in f32 too
    const int couple  = z0 / 9;
    const int c0      = z0 - couple * 9; // in {0,2,4,6}
    const int chlo    = (int)floorf((float)(c0 * 81) * R);
    const int chlo_al = chlo & ~3;       // 16B-aligned window start
    const int g0lo    = half * 7;

    __shared__ __align__(16) float lds[TOK_ROWS * CH_W];   // 36,064 bytes

    const float* src = hs + (size_t)b * 3137u * 768u;
    const int tokbase = couple * 196 + g0lo * 14 + 1;      // +1 skips CLS token

    // ---- Phase 1: async-stage 98 tokens x 92 channels into LDS (b128) ----
    for (int i = threadIdx.x; i < TOT_VEC; i += 256) {
        const int r = i / VEC_PER_ROW;           // token row    [0,98)
        const int q = i - r * VEC_PER_ROW;       // 16B chunk    [0,23)
        const float* gp = src + (size_t)(tokbase + r) * 768u
                              + (size_t)(chlo_al + q * 4);
        const uint64_t ga = (uint64_t)(size_t)gp;
        const uint32_t la = (uint32_t)(size_t)(&lds[r * CH_W + q * 4]);
        // CDNA5 async load: copies 16 bytes from the global address in ga to the LDS address in la, tracked by ASYNCcnt
        asm volatile("global_load_async_to_lds_b128 %0, %1, off"
                     :: "v"(la), "v"(ga) : "memory");
    }
    asm volatile("s_wait_asynccnt 0" ::: "memory");
    __syncthreads();

    // ---- Phase 2: compute 32x64 outputs from LDS, coalesced stores ----
    // idx = tid + k*256; since 256 % 64 == 0, o2 is constant per thread.
    const int o2  = threadIdx.x & 63;
    const int z2  = (o2 * 63) >> 5;              // floor(o2*126/64), exact
    const int g1  = z2 / 9;
    const int c2  = z2 - g1 * 9;
    const int o1b = (half << 5) + (threadIdx.x >> 6);   // o1 for k=0; +4 per k

    const size_t obase = (((size_t)b * 64 + o0) * 64) * 64;

    #pragma unroll
    for (int k = 0; k < 8; ++k) {
        const int o1 = o1b + k * 4;

        const int z1 = (o1 * 63) >> 5;           // floor(o1*126/64), exact
        const int g0 = z1 / 9;
        const int c1 = z1 - g0 * 9;

        const int ch729 = c0 * 81 + c1 * 9 + c2;
        // replicate JAX float32 rounding exactly (k=243 -> 255, k=486 -> 511)
        const int ch768 = (int)floorf((float)ch729 * R);

        const int row = (g0 - g0lo) * 14 + g1;
        const float v = lds[row * CH_W + (ch768 - chlo_al)];

        out[obase + (size_t)o1 * 64 + o2] = v;
    }
}

extern "C" void kernel_launch(void* const* d_in, const int* in_sizes, int n_in,
                              void* d_out, int out_size, void* d_ws, size_t ws_size,
                              hipStream_t stream) {
    (void)in_sizes; (void)n_in; (void)d_ws; (void)ws_size; (void)out_size;
    const float* hs = (const float*)d_in[0];
    float* out = (float*)d_out;
    dim3 grid(64 * 64 * 2);   // (b, o0, o1-half)
    dim3 block(256);
    FRAMES_VisionTransformer_28166395527587_kernel<<<grid, block, 0, stream>>>(hs, out);
}